// GCN_18047452578507
// MI455X (gfx1250) — compile-verified
//
#include <hip/hip_runtime.h>

typedef __attribute__((ext_vector_type(2))) float v2f;
typedef __attribute__((ext_vector_type(8))) float v8f;

// ---------------------------------------------------------------------------
// Degree / normalization kernels
// ---------------------------------------------------------------------------
__global__ void init_deg_kernel(float* __restrict__ deg, int n_nodes) {
    int i = blockIdx.x * blockDim.x + threadIdx.x;
    if (i < n_nodes) deg[i] = 1.0f;  // self-loop contributes 1
}

__global__ void accum_deg_kernel(float* __restrict__ deg,
                                 const int* __restrict__ dst, int n_edges) {
    int e = blockIdx.x * blockDim.x + threadIdx.x;
    if (e < n_edges) atomicAdd(&deg[dst[e]], 1.0f);
}

__global__ void deg_to_dinv_kernel(float* __restrict__ deg, int n_nodes) {
    int i = blockIdx.x * blockDim.x + threadIdx.x;
    if (i < n_nodes) deg[i] = rsqrtf(deg[i]);  // deg >= 1 always (self-loops)
}

// ---------------------------------------------------------------------------
// WMMA f32 GEMM:  C[M x (NT*16)] = act(A[M x 128]) * B[128 x (NT*16)]
// One wave -> 16 rows x full output width. K stepped by 4 with
// V_WMMA_F32_16X16X4_F32. Block = 128 threads = 4 waves = 64 rows.
// A 16x4 f32 layout:  VGPR v, lane l: A[l%16][k + v + 2*(l/16)]
// B  4x16 f32 layout: VGPR v, lane l: B[k + v + 2*(l/16)][l%16]
// C 16x16 f32 layout: VGPR r, lane l: C[r + 8*(l/16)][l%16]
// ---------------------------------------------------------------------------
template <int NT, bool RELU_A>
__global__ __launch_bounds__(128) void gemm_wmma_kernel(
    const float* __restrict__ A, const float* __restrict__ B,
    float* __restrict__ C, int M) {
    constexpr int K = 128;
    constexpr int NCOLS = NT * 16;

    const int lane = threadIdx.x & 31;
    const int wave = threadIdx.x >> 5;
    const int m    = lane & 15;   // row-in-tile / col-in-tile
    const int hi   = lane >> 4;   // 0 or 1 (lane half)

    const int row_base = blockIdx.x * 64 + wave * 16;
    int arow = row_base + m;
    if (arow > M - 1) arow = M - 1;  // clamp tail loads (stores are guarded)
    const float* Arow = A + (size_t)arow * K;

    v8f acc[NT] = {};

    for (int k = 0; k < K; k += 4) {
        float a0 = Arow[k + 2 * hi + 0];
        float a1 = Arow[k + 2 * hi + 1];
        if (RELU_A) {
            a0 = fmaxf(a0, 0.0f);
            a1 = fmaxf(a1, 0.0f);
        }
        v2f a;
        a.x = a0;
        a.y = a1;
#pragma unroll
        for (int t = 0; t < NT; ++t) {
            const float* Bp = B + (size_t)(k + 2 * hi) * NCOLS + t * 16 + m;
            v2f b;
            b.x = Bp[0];
            b.y = Bp[NCOLS];
            acc[t] = __builtin_amdgcn_wmma_f32_16x16x4_f32(
                false, a, false, b, (short)0, acc[t], false, false);
        }
    }

#pragma unroll
    for (int r = 0; r < 8; ++r) {
        int row = row_base + r + 8 * hi;
        if (row < M) {
            float* Crow = C + (size_t)row * NCOLS;
#pragma unroll
            for (int t = 0; t < NT; ++t) Crow[t * 16 + m] = acc[t][r];
        }
    }
}

// ---------------------------------------------------------------------------
// out[n][c] = bias[c] + dinv[n]^2 * t[n][c]   (self-loop term + bias)
// One thread = 4 columns (float4).
// ---------------------------------------------------------------------------
template <int COLS>
__global__ void init_agg_kernel(float* __restrict__ out,
                                const float* __restrict__ t,
                                const float* __restrict__ bias,
                                const float* __restrict__ dinv, int n_nodes) {
    constexpr int PER = COLS / 4;
    int idx  = blockIdx.x * blockDim.x + threadIdx.x;
    int node = idx / PER;
    int c4   = (idx % PER) * 4;
    if (node >= n_nodes) return;
    float w = dinv[node];
    w = w * w;
    const float4 tv = *(const float4*)(t + (size_t)node * COLS + c4);
    const float4 bv = *(const float4*)(bias + c4);
    float4 o;
    o.x = fmaf(w, tv.x, bv.x);
    o.y = fmaf(w, tv.y, bv.y);
    o.z = fmaf(w, tv.z, bv.z);
    o.w = fmaf(w, tv.w, bv.w);
    *(float4*)(out + (size_t)node * COLS + c4) = o;
}

// ---------------------------------------------------------------------------
// Edge scatter-add: out[dst] += dinv[src]*dinv[dst] * t[src]
// COLS/4 threads per edge (COLS=128 -> one full wave per edge, so src/dst/
// dinv loads scalarize). Float atomics resolve in L2 (accumulator fits).
// ---------------------------------------------------------------------------
template <int COLS>
__global__ void edge_agg_kernel(float* __restrict__ out,
                                const float* __restrict__ t,
                                const int* __restrict__ src,
                                const int* __restrict__ dst,
                                const float* __restrict__ dinv, int n_edges) {
    constexpr int PER = COLS / 4;
    int idx = blockIdx.x * blockDim.x + threadIdx.x;
    int e   = idx / PER;
    int c4  = (idx % PER) * 4;
    if (e >= n_edges) return;
    int s = src[e];
    int d = dst[e];
    float w = dinv[s] * dinv[d];
    const float4 tv = *(const float4*)(t + (size_t)s * COLS + c4);
    float* o = out + (size_t)d * COLS + c4;
    atomicAdd(o + 0, w * tv.x);
    atomicAdd(o + 1, w * tv.y);
    atomicAdd(o + 2, w * tv.z);
    atomicAdd(o + 3, w * tv.w);
}

// ---------------------------------------------------------------------------
extern "C" void kernel_launch(void* const* d_in, const int* in_sizes, int n_in,
                              void* d_out, int out_size, void* d_ws,
                              size_t ws_size, hipStream_t stream) {
    const float* x  = (const float*)d_in[0];
    const int*   ei = (const int*)d_in[1];
    const float* W1 = (const float*)d_in[2];
    const float* b1 = (const float*)d_in[3];
    const float* W2 = (const float*)d_in[4];
    const float* b2 = (const float*)d_in[5];

    const int IN_DIM = 128, HID_DIM = 128, OUT_DIM = 64;
    const int n_nodes = in_sizes[0] / IN_DIM;
    const int n_edges = in_sizes[1] / 2;
    const int* src = ei;
    const int* dst = ei + n_edges;
    float* out = (float*)d_out;

    // Workspace layout (t2 reuses t1's buffer):
    char*  ws   = (char*)d_ws;
    size_t off  = 0;
    float* dinv = (float*)(ws + off);
    off += ((size_t)n_nodes * 4 + 255) & ~(size_t)255;
    float* t1 = (float*)(ws + off);  // x @ W1, [N,128]
    off += ((size_t)n_nodes * HID_DIM * 4 + 255) & ~(size_t)255;
    float* h = (float*)(ws + off);   // aggregated layer-1 (pre-ReLU), [N,128]
    float* t2 = t1;                  // h @ W2, [N,64] — reuse t1 storage

    const int B = 256;

    // 1) symmetric normalization  D^{-1/2}
    init_deg_kernel<<<(n_nodes + B - 1) / B, B, 0, stream>>>(dinv, n_nodes);
    accum_deg_kernel<<<(n_edges + B - 1) / B, B, 0, stream>>>(dinv, dst, n_edges);
    deg_to_dinv_kernel<<<(n_nodes + B - 1) / B, B, 0, stream>>>(dinv, n_nodes);

    // 2) t1 = x @ W1   (WMMA f32, 8 col-tiles of 16)
    gemm_wmma_kernel<8, false>
        <<<(n_nodes + 63) / 64, 128, 0, stream>>>(x, W1, t1, n_nodes);

    // 3) h = b1 + dinv^2*t1, then edge scatter-add
    {
        int threads = n_nodes * (HID_DIM / 4);
        init_agg_kernel<128>
            <<<(threads + B - 1) / B, B, 0, stream>>>(h, t1, b1, dinv, n_nodes);
        int ethreads = n_edges * (HID_DIM / 4);
        edge_agg_kernel<128><<<(ethreads + B - 1) / B, B, 0, stream>>>(
            h, t1, src, dst, dinv, n_edges);
    }

    // 4) t2 = relu(h) @ W2   (WMMA f32, ReLU fused on A-load, 4 col-tiles)
    gemm_wmma_kernel<4, true>
        <<<(n_nodes + 63) / 64, 128, 0, stream>>>(h, W2, t2, n_nodes);

    // 5) out = b2 + dinv^2*t2, then edge scatter-add into d_out
    {
        int threads = n_nodes * (OUT_DIM / 4);
        init_agg_kernel<64>
            <<<(threads + B - 1) / B, B, 0, stream>>>(out, t2, b2, dinv, n_nodes);
        int ethreads = n_edges * (OUT_DIM / 4);
        edge_agg_kernel<64><<<(ethreads + B - 1) / B, B, 0, stream>>>(
            out, t2, src, dst, dinv, n_edges);
    }
}